// SharedCache_54288386621471
// MI455X (gfx1250) — compile-verified
//
#include <hip/hip_runtime.h>
#include <stdint.h>

namespace {
constexpr int    kB          = 8;
constexpr int    kS          = 4096;
constexpr int    kH          = 8;
constexpr int    kD          = 128;
constexpr int    kRowFloats  = kH * kD;                        // 1024 floats = 4 KB per row
constexpr int    kMaxTotal   = kB * kS;                        // 32768 packed rows
constexpr size_t kCacheElems = (size_t)kMaxTotal * kRowFloats; // 33,554,432 floats per cache
constexpr int    kThreads    = 256;                            // one float4 per thread per row
constexpr int    kRowsPerBlock = 4;                            // 4 K-rows + 4 V-rows per block
}

__global__ __launch_bounds__(kThreads)
void pack_kv_async_kernel(const float* __restrict__ key,
                          const float* __restrict__ val,
                          const int*   __restrict__ seq_lens,
                          float*       __restrict__ out)
{
    __shared__ int    s_cum[kB + 1];
    __shared__ float4 s_stage[kRowsPerBlock * 2 * kThreads];   // 32 KB staging

    const int tid = threadIdx.x;

    // Tiny per-block cumsum (B == 8): thread 0 computes, all threads cache it.
    if (tid == 0) {
        int c = 0;
        s_cum[0] = 0;
#pragma unroll
        for (int b = 0; b < kB; ++b) { c += seq_lens[b]; s_cum[b + 1] = c; }
    }
    __syncthreads();

    int cum[kB + 1];
#pragma unroll
    for (int i = 0; i <= kB; ++i) cum[i] = s_cum[i];
    const int total = cum[kB];                                  // <= 8*4095 < 32768

    float* out_k = out;
    float* out_v = out + kCacheElems;

    // Metadata tail: seq_lens (8) then cumsum (9), numeric-cast to f32.
    if (blockIdx.x == 0) {
        if (tid < kB)     out[2 * kCacheElems + tid]      = (float)seq_lens[tid];
        if (tid < kB + 1) out[2 * kCacheElems + kB + tid] = (float)cum[tid];
    }

    const int      row0     = blockIdx.x * kRowsPerBlock;
    // Workgroup-relative LDS byte offset = low 32 bits of the generic address.
    const uint32_t lds_base = (uint32_t)(uintptr_t)&s_stage[0];

    // ---- Phase 1: async DMA gather of valid source rows into LDS staging ----
#pragma unroll
    for (int r = 0; r < kRowsPerBlock; ++r) {
        const int row = row0 + r;                // uniform per block -> uniform branch
        if (row < total) {
            // largest b with cum[b] <= row  (handles zero-length batches)
            int b = 0;
#pragma unroll
            for (int i = 0; i < kB - 1; ++i) if (row >= cum[i + 1]) b = i + 1;
            const int    t   = row - cum[b];
            const size_t src = ((size_t)b * kS + t) * kRowFloats + tid * 4;
            const uint64_t ka = (uint64_t)(uintptr_t)(key + src);
            const uint64_t va = (uint64_t)(uintptr_t)(val + src);
            const uint32_t lk = lds_base + (uint32_t)((r * 2 + 0) * kThreads + tid) * 16u;
            const uint32_t lv = lds_base + (uint32_t)((r * 2 + 1) * kThreads + tid) * 16u;
            asm volatile("global_load_async_to_lds_b128 %0, %1, off"
                         :: "v"(lk), "v"(ka) : "memory");
            asm volatile("global_load_async_to_lds_b128 %0, %1, off"
                         :: "v"(lv), "v"(va) : "memory");
        }
    }
    // Each wave only consumes LDS written by its own async loads.
    asm volatile("s_wait_asynccnt 0x0" ::: "memory");

    // ---- Phase 2: async DMA store of packed rows; zero-fill rows >= total ----
#pragma unroll
    for (int r = 0; r < kRowsPerBlock; ++r) {
        const int    row = row0 + r;
        const size_t dst = (size_t)row * kRowFloats + tid * 4;
        if (row < total) {
            const uint64_t ka = (uint64_t)(uintptr_t)(out_k + dst);
            const uint64_t va = (uint64_t)(uintptr_t)(out_v + dst);
            const uint32_t lk = lds_base + (uint32_t)((r * 2 + 0) * kThreads + tid) * 16u;
            const uint32_t lv = lds_base + (uint32_t)((r * 2 + 1) * kThreads + tid) * 16u;
            asm volatile("global_store_async_from_lds_b128 %0, %1, off"
                         :: "v"(ka), "v"(lk) : "memory");
            asm volatile("global_store_async_from_lds_b128 %0, %1, off"
                         :: "v"(va), "v"(lv) : "memory");
        } else {
            const float4 z = make_float4(0.f, 0.f, 0.f, 0.f);
            *(float4*)(out_k + dst) = z;   // global_store_b128
            *(float4*)(out_v + dst) = z;
        }
    }
    asm volatile("s_wait_asynccnt 0x0" ::: "memory");   // stores complete before endpgm
}

extern "C" void kernel_launch(void* const* d_in, const int* in_sizes, int n_in,
                              void* d_out, int out_size, void* d_ws, size_t ws_size,
                              hipStream_t stream) {
    (void)in_sizes; (void)n_in; (void)out_size; (void)d_ws; (void)ws_size;
    const float* key      = (const float*)d_in[0];
    const float* val      = (const float*)d_in[1];
    const int*   seq_lens = (const int*)d_in[2];
    float*       out      = (float*)d_out;

    const int grid = kMaxTotal / kRowsPerBlock;   // 8192 blocks x 256 threads (8 waves)
    pack_kv_async_kernel<<<grid, kThreads, 0, stream>>>(key, val, seq_lens, out);
}